// DCN_70523363000935
// MI455X (gfx1250) — compile-verified
//
#include <hip/hip_runtime.h>
#include <hip/hip_bf16.h>

// ---------------- problem constants (from reference) ----------------
#define N_ROWS 131072
#define DIM    64
#define K_CL   1024
#define ITERS  10
#define GAMMA  0.01f
#define EPSV   1e-10f

// ---------------- tiling ----------------
#define BM   128          // rows per block tile
#define BN   128          // cluster cols per block tile
#define LDX  68           // padded LDS stride for X tile (conflict-free, 16B aligned rows)
#define LDC  68           // padded LDS stride for C tile
#define LDSP 132          // padded LDS stride for sp tile
#define P3_YB 32          // pass-3 row groups (partials summed deterministically)
#define P3_CHUNKS (N_ROWS / P3_YB / BM)   // 32 chunks of 128 rows per pass-3 block

typedef float v2f __attribute__((ext_vector_type(2)));
typedef float v8f __attribute__((ext_vector_type(8)));

// CDNA5 fp32 WMMA: D(16x16) = A(16x4) * B(4x16) + C   (exact fp32 path)
__device__ __forceinline__ v8f wmma_k4(v2f a, v2f b, v8f c) {
  return __builtin_amdgcn_wmma_f32_16x16x4_f32(false, a, false, b, (short)0, c, false, false);
}

// Full K=64 dot tile with double-buffered B fragments: two 4-frag buffers with
// overlapping live ranges force the allocator to keep both resident, so each
// 4-wmma group waits only on its own (already prefetched) loads instead of a
// full dscnt drain per wmma.
__device__ __forceinline__ v8f gemm16_k64(const float* __restrict__ xp,
                                          const float* __restrict__ cp) {
  v2f af[16];
#pragma unroll
  for (int k4 = 0; k4 < 16; ++k4) {
    af[k4].x = xp[k4 * 4];
    af[k4].y = xp[k4 * 4 + 1];
  }
  v2f bA[4], bB[4];
#pragma unroll
  for (int q = 0; q < 4; ++q) { bA[q].x = cp[q * 4];        bA[q].y = cp[q * 4 + 1]; }
#pragma unroll
  for (int q = 0; q < 4; ++q) { bB[q].x = cp[(4 + q) * 4];  bB[q].y = cp[(4 + q) * 4 + 1]; }

  v8f acc = {};
#pragma unroll
  for (int g = 0; g < 4; ++g) {
    if ((g & 1) == 0) {
      // consume bA, prefetch group g+2 into bA afterwards via bB ping-pong
#pragma unroll
      for (int q = 0; q < 4; ++q) acc = wmma_k4(af[g * 4 + q], bA[q], acc);
      if (g + 2 < 4) {
#pragma unroll
        for (int q = 0; q < 4; ++q) {
          bA[q].x = cp[((g + 2) * 4 + q) * 4];
          bA[q].y = cp[((g + 2) * 4 + q) * 4 + 1];
        }
      }
    } else {
#pragma unroll
      for (int q = 0; q < 4; ++q) acc = wmma_k4(af[g * 4 + q], bB[q], acc);
      if (g + 2 < 4) {
#pragma unroll
        for (int q = 0; q < 4; ++q) {
          bB[q].x = cp[((g + 2) * 4 + q) * 4];
          bB[q].y = cp[((g + 2) * 4 + q) * 4 + 1];
        }
      }
    }
  }
  return acc;
}

// order-preserving float<->uint map so min/max atomics are associative+deterministic
__device__ __forceinline__ unsigned f2ord(float f) {
  unsigned u = __float_as_uint(f);
  return (u & 0x80000000u) ? ~u : (u | 0x80000000u);
}
__device__ __forceinline__ float ord2f(unsigned o) {
  return (o & 0x80000000u) ? __uint_as_float(o ^ 0x80000000u) : __uint_as_float(~o);
}
__device__ __forceinline__ unsigned umin2(unsigned a, unsigned b) { return a < b ? a : b; }
__device__ __forceinline__ unsigned umax2(unsigned a, unsigned b) { return a > b ? a : b; }

// ---------------- K0: ||x||^2 per row (once) ----------------
__global__ void k_xnorm(const float* __restrict__ X, float* __restrict__ xnorm) {
  int t = blockIdx.x * blockDim.x + threadIdx.x;
  if (t >= N_ROWS) return;
  const float4* xr = reinterpret_cast<const float4*>(X + (size_t)t * DIM);
  float s = 0.f;
#pragma unroll
  for (int i = 0; i < DIM / 4; ++i) {
    float4 v = xr[i];
    s += v.x * v.x + v.y * v.y + v.z * v.z + v.w * v.w;
  }
  xnorm[t] = s;
}

// ---------------- K1: ||c||^2 + min/max reset (per iteration) ----------------
__global__ void k_reset(const float* __restrict__ C, float* __restrict__ cnorm,
                        unsigned* __restrict__ mmu) {
  int t = blockIdx.x * blockDim.x + threadIdx.x;
  if (t < K_CL) {
    const float4* cr = reinterpret_cast<const float4*>(C + (size_t)t * DIM);
    float s = 0.f;
#pragma unroll
    for (int i = 0; i < DIM / 4; ++i) {
      float4 v = cr[i];
      s += v.x * v.x + v.y * v.y + v.z * v.z + v.w * v.w;
    }
    cnorm[t] = s;
  }
  if (t == 0) {
    mmu[0] = f2ord(__builtin_inff());    // running min (ordered space)
    mmu[1] = f2ord(-__builtin_inff());   // running max (ordered space)
  }
}

// ---------------- K2: global min/max of d via WMMA tiles ----------------
__global__ void k_pass1(const float* __restrict__ X, const float* __restrict__ C,
                        const float* __restrict__ xnorm, const float* __restrict__ cnorm,
                        unsigned* __restrict__ mmu) {
  extern __shared__ float smem[];
  float* Xs = smem;                  // BM*LDX
  float* Cs = Xs + BM * LDX;         // BN*LDC
  float* XN = Cs + BN * LDC;         // BM
  float* CN = XN + BM;               // BN
  unsigned* red = reinterpret_cast<unsigned*>(CN + BN);  // 16

  const int tid = threadIdx.x, lane = tid & 31, w = tid >> 5;
  const int row0 = blockIdx.x * BM, col0 = blockIdx.y * BN;

  for (int idx = tid; idx < BM * (DIM / 4); idx += 256) {
    int r = idx >> 4, c = (idx & 15) << 2;
    *reinterpret_cast<float4*>(&Xs[r * LDX + c]) =
        *reinterpret_cast<const float4*>(X + (size_t)(row0 + r) * DIM + c);
  }
  for (int idx = tid; idx < BN * (DIM / 4); idx += 256) {
    int r = idx >> 4, c = (idx & 15) << 2;
    *reinterpret_cast<float4*>(&Cs[r * LDC + c]) =
        *reinterpret_cast<const float4*>(C + (size_t)(col0 + r) * DIM + c);
  }
  if (tid < BM) XN[tid] = xnorm[row0 + tid];
  else          CN[tid - BM] = cnorm[col0 + tid - BM];
  __syncthreads();

  const int strip = w << 4;
  const int hb = (lane < 16) ? 0 : 8;   // C/D layout: high half-wave holds M+8
  const int lc = lane & 15;
  const int ko = (lane >> 4) << 1;      // A/B layout: high half-wave holds K+2
  float xnr[8];
#pragma unroll
  for (int i = 0; i < 8; ++i) xnr[i] = XN[strip + hb + i];
  const float* xp = &Xs[(strip + lc) * LDX + ko];

  unsigned lmin = 0xFFFFFFFFu, lmax = 0u;
  for (int ct = 0; ct < 8; ++ct) {
    const float* cp = &Cs[((ct << 4) + lc) * LDC + ko];
    v8f acc = gemm16_k64(xp, cp);
    float cn = CN[(ct << 4) + lc];
#pragma unroll
    for (int i = 0; i < 8; ++i) {
      float d = __builtin_fmaf(-2.f, acc[i], xnr[i] + cn);
      unsigned o = f2ord(d);
      lmin = umin2(lmin, o);
      lmax = umax2(lmax, o);
    }
  }
#pragma unroll
  for (int m = 16; m >= 1; m >>= 1) {
    lmin = umin2(lmin, (unsigned)__shfl_xor((int)lmin, m, 32));
    lmax = umax2(lmax, (unsigned)__shfl_xor((int)lmax, m, 32));
  }
  if (lane == 0) { red[w] = lmin; red[8 + w] = lmax; }
  __syncthreads();
  if (tid == 0) {
    unsigned bmin = red[0], bmax = red[8];
#pragma unroll
    for (int i = 1; i < 8; ++i) { bmin = umin2(bmin, red[i]); bmax = umax2(bmax, red[8 + i]); }
    atomicMin(&mmu[0], bmin);
    atomicMax(&mmu[1], bmax);
  }
}

// ---------------- K3: rowsum(soft) -- block owns 128 rows x all K, deterministic ----------------
__global__ void k_pass2(const float* __restrict__ X, const float* __restrict__ C,
                        const float* __restrict__ xnorm, const float* __restrict__ cnorm,
                        const unsigned* __restrict__ mmu, float* __restrict__ rowsum) {
  extern __shared__ float smem[];
  float* Xs = smem;
  float* Cs = Xs + BM * LDX;
  float* XN = Cs + BN * LDC;
  float* CN = XN + BM;

  const int tid = threadIdx.x, lane = tid & 31, w = tid >> 5;
  const int row0 = blockIdx.x * BM;

  for (int idx = tid; idx < BM * (DIM / 4); idx += 256) {
    int r = idx >> 4, c = (idx & 15) << 2;
    *reinterpret_cast<float4*>(&Xs[r * LDX + c]) =
        *reinterpret_cast<const float4*>(X + (size_t)(row0 + r) * DIM + c);
  }
  if (tid < BM) XN[tid] = xnorm[row0 + tid];
  __syncthreads();

  const float dmin = ord2f(mmu[0]);
  const float dmax = ord2f(mmu[1]);
  const float scale = -GAMMA / (dmax - dmin);

  const int strip = w << 4, hb = (lane < 16) ? 0 : 8, lc = lane & 15, ko = (lane >> 4) << 1;
  float xnr[8];
#pragma unroll
  for (int i = 0; i < 8; ++i) xnr[i] = XN[strip + hb + i];
  const float* xp = &Xs[(strip + lc) * LDX + ko];

  float rowacc[8] = {0, 0, 0, 0, 0, 0, 0, 0};

  for (int cb = 0; cb < K_CL / BN; ++cb) {
    __syncthreads();  // previous chunk's compute done before Cs reload
    const int col0 = cb * BN;
    for (int idx = tid; idx < BN * (DIM / 4); idx += 256) {
      int r = idx >> 4, c = (idx & 15) << 2;
      *reinterpret_cast<float4*>(&Cs[r * LDC + c]) =
          *reinterpret_cast<const float4*>(C + (size_t)(col0 + r) * DIM + c);
    }
    if (tid < BN) CN[tid] = cnorm[col0 + tid];
    __syncthreads();

    for (int ct = 0; ct < 8; ++ct) {
      const float* cp = &Cs[((ct << 4) + lc) * LDC + ko];
      v8f acc = gemm16_k64(xp, cp);
      float cn = CN[(ct << 4) + lc];
#pragma unroll
      for (int i = 0; i < 8; ++i) {
        float d = __builtin_fmaf(-2.f, acc[i], xnr[i] + cn);
        rowacc[i] += __expf((d - dmin) * scale);
      }
    }
  }
  // reduce across the 16 column-lanes of each half-wave; each row written by exactly one lane
#pragma unroll
  for (int i = 0; i < 8; ++i) {
    float v = rowacc[i];
#pragma unroll
    for (int m = 1; m <= 8; m <<= 1) v += __shfl_xor(v, m, 32);
    if (lc == 0) rowsum[row0 + strip + hb + i] = v;
  }
}

// ---------------- K4: accumulate sp^T @ X and colsum(sp) into per-block partials ----------------
__global__ void k_pass3(const float* __restrict__ X, const float* __restrict__ C,
                        const float* __restrict__ xnorm, const float* __restrict__ cnorm,
                        const unsigned* __restrict__ mmu, const float* __restrict__ rowsum,
                        float* __restrict__ pacc, float* __restrict__ pcol) {
  extern __shared__ float smem[];
  float* Xs  = smem;                 // BM*LDX
  float* Cs  = Xs + BM * LDX;        // BN*LDC
  float* SPs = Cs + BN * LDC;        // BM*LDSP
  float* CN  = SPs + BM * LDSP;      // BN
  float* XN  = CN + BN;              // BM
  float* IRS = XN + BM;              // BM (1/rowsum)

  const int tid = threadIdx.x, lane = tid & 31, w = tid >> 5;
  const int cb = blockIdx.x;                 // cluster block 0..7
  const int yb = blockIdx.y;                 // row group 0..P3_YB-1
  const int col0 = cb * BN;
  const int rbase = yb * (N_ROWS / P3_YB);

  for (int idx = tid; idx < BN * (DIM / 4); idx += 256) {
    int r = idx >> 4, c = (idx & 15) << 2;
    *reinterpret_cast<float4*>(&Cs[r * LDC + c]) =
        *reinterpret_cast<const float4*>(C + (size_t)(col0 + r) * DIM + c);
  }
  if (tid < BN) CN[tid] = cnorm[col0 + tid];

  const float dmin = ord2f(mmu[0]);
  const float dmax = ord2f(mmu[1]);
  const float scale = -GAMMA / (dmax - dmin);

  const int strip = w << 4, hb = (lane < 16) ? 0 : 8, lc = lane & 15, ko = (lane >> 4) << 1;

  v8f acc2[4];
#pragma unroll
  for (int nt = 0; nt < 4; ++nt) acc2[nt] = (v8f){};
  float csum = 0.f;

  for (int ch = 0; ch < P3_CHUNKS; ++ch) {
    const int row0 = rbase + ch * BM;
    __syncthreads();  // previous chunk's SPs/Xs consumers done (also covers Cs/CN first use)
    for (int idx = tid; idx < BM * (DIM / 4); idx += 256) {
      int r = idx >> 4, c = (idx & 15) << 2;
      *reinterpret_cast<float4*>(&Xs[r * LDX + c]) =
          *reinterpret_cast<const float4*>(X + (size_t)(row0 + r) * DIM + c);
    }
    if (tid < BM) XN[tid] = xnorm[row0 + tid];
    else          IRS[tid - BM] = 1.0f / rowsum[row0 + tid - BM];
    __syncthreads();

    // ---- GEMM1: d -> sp, transposed into LDS ----
    float xnr[8], irsr[8];
#pragma unroll
    for (int i = 0; i < 8; ++i) {
      xnr[i]  = XN[strip + hb + i];
      irsr[i] = IRS[strip + hb + i];
    }
    const float* xp = &Xs[(strip + lc) * LDX + ko];
    for (int ct = 0; ct < 8; ++ct) {
      const float* cp = &Cs[((ct << 4) + lc) * LDC + ko];
      v8f acc = gemm16_k64(xp, cp);
      float cn = CN[(ct << 4) + lc];
#pragma unroll
      for (int i = 0; i < 8; ++i) {
        float d  = __builtin_fmaf(-2.f, acc[i], xnr[i] + cn);
        float sp = __expf((d - dmin) * scale) * irsr[i] + EPSV;
        SPs[(strip + hb + i) * LDSP + (ct << 4) + lc] = sp;
      }
    }
    __syncthreads();

    // ---- GEMM2: acc2 += sp^T(clusters x rows) @ X(rows x 64), wave owns 16 clusters.
    // Ping-pong staging across the 8 k-groups: group g+1's 20 ds_loads are in
    // flight while group g's 16 wmmas execute.
    const float* spA = &SPs[(w << 4) + lc];
    const float* xB0 = &Xs[lc];
    v2f a0[4], b0[4][4], a1[4], b1[4][4];
#pragma unroll
    for (int q = 0; q < 4; ++q) {
      int k0 = q * 4 + ko;
      a0[q].x = spA[k0 * LDSP];
      a0[q].y = spA[(k0 + 1) * LDSP];
#pragma unroll
      for (int nt = 0; nt < 4; ++nt) {
        b0[q][nt].x = xB0[k0 * LDX + (nt << 4)];
        b0[q][nt].y = xB0[(k0 + 1) * LDX + (nt << 4)];
      }
    }
#pragma unroll
    for (int g = 0; g < 8; ++g) {
      if ((g & 1) == 0) {
        if (g + 1 < 8) {
#pragma unroll
          for (int q = 0; q < 4; ++q) {
            int k0 = (g + 1) * 16 + q * 4 + ko;
            a1[q].x = spA[k0 * LDSP];
            a1[q].y = spA[(k0 + 1) * LDSP];
#pragma unroll
            for (int nt = 0; nt < 4; ++nt) {
              b1[q][nt].x = xB0[k0 * LDX + (nt << 4)];
              b1[q][nt].y = xB0[(k0 + 1) * LDX + (nt << 4)];
            }
          }
        }
#pragma unroll
        for (int q = 0; q < 4; ++q)
#pragma unroll
          for (int nt = 0; nt < 4; ++nt) acc2[nt] = wmma_k4(a0[q], b0[q][nt], acc2[nt]);
      } else {
        if (g + 1 < 8) {
#pragma unroll
          for (int q = 0; q < 4; ++q) {
            int k0 = (g + 1) * 16 + q * 4 + ko;
            a0[q].x = spA[k0 * LDSP];
            a0[q].y = spA[(k0 + 1) * LDSP];
#pragma unroll
            for (int nt = 0; nt < 4; ++nt) {
              b0[q][nt].x = xB0[k0 * LDX + (nt << 4)];
              b0[q][nt].y = xB0[(k0 + 1) * LDX + (nt << 4)];
            }
          }
        }
#pragma unroll
        for (int q = 0; q < 4; ++q)
#pragma unroll
          for (int nt = 0; nt < 4; ++nt) acc2[nt] = wmma_k4(a1[q], b1[q][nt], acc2[nt]);
      }
    }
    // ---- colsum partial (thread t owns cluster column t) ----
    if (tid < BN) {
      float s = 0.f;
      for (int r = 0; r < BM; ++r) s += SPs[r * LDSP + tid];
      csum += s;
    }
  }

  // plain stores into this block's private partial slice (deterministic; no atomics)
#pragma unroll
  for (int nt = 0; nt < 4; ++nt)
#pragma unroll
    for (int i = 0; i < 8; ++i) {
      int cl = col0 + (w << 4) + hb + i;
      pacc[(size_t)yb * (K_CL * DIM) + (size_t)cl * DIM + (nt << 4) + lc] = acc2[nt][i];
    }
  if (tid < BN) pcol[yb * K_CL + col0 + tid] = csum;
}

// ---------------- K5: newC = sum(partials)/colsum + eps, fixed reduction order ----------------
__global__ void k_finalize(const float* __restrict__ pacc, const float* __restrict__ pcol,
                           float* __restrict__ Cout) {
  int t = blockIdx.x * blockDim.x + threadIdx.x;
  if (t >= K_CL * DIM) return;
  int k = t >> 6;
  float s = 0.f, cs = 0.f;
  for (int p = 0; p < P3_YB; ++p) {
    s  += pacc[(size_t)p * (K_CL * DIM) + t];
    cs += pcol[p * K_CL + k];
  }
  Cout[t] = s / cs + EPSV;
}

extern "C" void kernel_launch(void* const* d_in, const int* in_sizes, int n_in,
                              void* d_out, int out_size, void* d_ws, size_t ws_size,
                              hipStream_t stream) {
  (void)in_sizes; (void)n_in; (void)out_size; (void)ws_size;
  const float* X  = (const float*)d_in[0];
  const float* C0 = (const float*)d_in[1];
  float* Cbuf = (float*)d_out;  // evolving clusters live in d_out

  // workspace layout (~9.6 MB)
  float* ws     = (float*)d_ws;
  float* xnorm  = ws;  ws += N_ROWS;
  float* rowsum = ws;  ws += N_ROWS;
  float* cnorm  = ws;  ws += K_CL;
  float* pcol   = ws;  ws += (size_t)P3_YB * K_CL;
  unsigned* mmu = (unsigned*)ws;  ws += 2;
  float* pacc   = ws;  ws += (size_t)P3_YB * K_CL * DIM;

  hipMemcpyAsync(Cbuf, C0, sizeof(float) * K_CL * DIM, hipMemcpyDeviceToDevice, stream);
  k_xnorm<<<N_ROWS / 256, 256, 0, stream>>>(X, xnorm);

  const size_t sm12 = (size_t)(BM * LDX + BN * LDC + BM + BN + 16) * sizeof(float);
  const size_t sm3  = (size_t)(BM * LDX + BN * LDC + BM * LDSP + BN + BM + BM) * sizeof(float);

  for (int it = 0; it < ITERS; ++it) {
    k_reset<<<K_CL / 256, 256, 0, stream>>>(Cbuf, cnorm, mmu);
    k_pass1<<<dim3(N_ROWS / BM, K_CL / BN), 256, sm12, stream>>>(X, Cbuf, xnorm, cnorm, mmu);
    k_pass2<<<N_ROWS / BM, 256, sm12, stream>>>(X, Cbuf, xnorm, cnorm, mmu, rowsum);
    k_pass3<<<dim3(K_CL / BN, P3_YB), 256, sm3, stream>>>(X, Cbuf, xnorm, cnorm, mmu, rowsum,
                                                          pacc, pcol);
    k_finalize<<<(K_CL * DIM) / 256, 256, 0, stream>>>(pacc, pcol, Cbuf);
  }
}